// Summary_77996606095382
// MI455X (gfx1250) — compile-verified
//
#include <hip/hip_runtime.h>

typedef float v2f __attribute__((ext_vector_type(2)));
typedef float v8f __attribute__((ext_vector_type(8)));

#define INO 128
#define INN 128
#define HID 64
#define BB  512
#define NN  2048

// ---------------------------------------------------------------------------
// Kernel A: fused projection GEMM via v_wmma_f32_16x16x4_f32 (full f32).
// Rows 0..511   : s = xo @ W1[:128] + b1   (b1 folded into the xo side only)
// Rows 512..2559: t = xn @ W1[128:]
// One wave (32 threads) per 16x16 output tile, K=128 -> 32 WMMAs.
// ---------------------------------------------------------------------------
__global__ __launch_bounds__(32) void proj_wmma(const float* __restrict__ xo,
                                                const float* __restrict__ xn,
                                                const float* __restrict__ W1,
                                                const float* __restrict__ b1,
                                                float* __restrict__ st) {
  const int rowTile = blockIdx.x;   // 0..159  (2560 rows / 16)
  const int colTile = blockIdx.y;   // 0..3    (64 cols / 16)
  const int lane = threadIdx.x;
  const int half = lane >> 4;       // 0/1
  const int l16  = lane & 15;
  const int rowBase = rowTile * 16;
  const int colBase = colTile * 16;

  const bool is_xo  = rowBase < BB;
  const float* Asrc = is_xo ? (xo + rowBase * INO) : (xn + (rowBase - BB) * INN);
  const int    wOff = is_xo ? 0 : INO;   // which half of W1

  v8f acc = {};
  for (int k = 0; k < INO; k += 4) {
    // A fragment: lane holds A[m=l16][k + 2*half + {0,1}]  (contiguous pair)
    const float* ap = Asrc + l16 * INO + k + 2 * half;
    v2f a; a.x = ap[0]; a.y = ap[1];
    // B fragment: lane holds W1[wOff + k + 2*half + r][colBase + l16]
    v2f b;
    b.x = W1[(wOff + k + 2 * half + 0) * HID + colBase + l16];
    b.y = W1[(wOff + k + 2 * half + 1) * HID + colBase + l16];
    acc = __builtin_amdgcn_wmma_f32_16x16x4_f32(false, a, false, b,
                                                (short)0, acc, false, false);
  }
  // D element [r + 8*half][l16] of the tile
  #pragma unroll
  for (int r = 0; r < 8; ++r) {
    const int m = rowBase + r + 8 * half;
    float v = acc[r];
    if (m < BB) v += b1[colBase + l16];   // z = s_b + t_n + b1: fold b1 once
    st[m * HID + colBase + l16] = v;
  }
}

// ---------------------------------------------------------------------------
// Kernel B: per-row dot with w2.  u[b] = w2 . s_b  (rows < 512),
//                                 v[n] = w2 . t_n  (rows >= 512)
// ---------------------------------------------------------------------------
__global__ __launch_bounds__(256) void rowdot_w2(const float* __restrict__ st,
                                                 const float* __restrict__ W2,
                                                 float* __restrict__ uv) {
  const int row = blockIdx.x * blockDim.x + threadIdx.x;
  if (row >= BB + NN) return;
  const float* p = st + row * HID;
  float acc = 0.0f;
  #pragma unroll
  for (int h = 0; h < HID; ++h) acc = fmaf(W2[h], p[h], acc);
  uv[row] = acc;
}

// ---------------------------------------------------------------------------
// Kernel C: one workgroup per b.
// scores[b,n] = 0.6*(u_b + v_n) + 0.4 * sum_h w2[h]*|s_b[h] + t_n[h]|
//   (leaky_relu(z,0.2) = 0.6 z + 0.4 |z|; b2 is softmax-invariant, dropped)
// Two-pass softmax in LDS, store normalized att row to workspace.
// ---------------------------------------------------------------------------
__global__ __launch_bounds__(256) void scores_softmax(
    const float* __restrict__ st, const float* __restrict__ uv,
    const float* __restrict__ W2, float* __restrict__ att) {
  const int b   = blockIdx.x;
  const int tid = threadIdx.x;

  __shared__ float s_s[HID];
  __shared__ float w2s[HID];
  __shared__ float sc[NN];
  __shared__ float red[256];

  if (tid < HID) { s_s[tid] = st[b * HID + tid]; w2s[tid] = W2[tid]; }
  __syncthreads();

  const float u_b = uv[b];
  float lmax = -3.0e38f;
  #pragma unroll
  for (int i = 0; i < NN / 256; ++i) {
    const int n = tid + 256 * i;
    const float4* tp = (const float4*)(st + (BB + n) * HID);  // 8B/16B aligned
    float acc = 0.0f;
    #pragma unroll
    for (int h4 = 0; h4 < HID / 4; ++h4) {
      const float4 t = tp[h4];
      acc = fmaf(w2s[4 * h4 + 0], fabsf(s_s[4 * h4 + 0] + t.x), acc);
      acc = fmaf(w2s[4 * h4 + 1], fabsf(s_s[4 * h4 + 1] + t.y), acc);
      acc = fmaf(w2s[4 * h4 + 2], fabsf(s_s[4 * h4 + 2] + t.z), acc);
      acc = fmaf(w2s[4 * h4 + 3], fabsf(s_s[4 * h4 + 3] + t.w), acc);
    }
    const float sc_n = 0.6f * (u_b + uv[BB + n]) + 0.4f * acc;
    sc[n] = sc_n;
    lmax = fmaxf(lmax, sc_n);
  }

  red[tid] = lmax; __syncthreads();
  for (int s = 128; s > 0; s >>= 1) {
    if (tid < s) red[tid] = fmaxf(red[tid], red[tid + s]);
    __syncthreads();
  }
  const float gmax = red[0];
  __syncthreads();

  float lsum = 0.0f;
  #pragma unroll
  for (int i = 0; i < NN / 256; ++i) {
    const int n = tid + 256 * i;
    const float p = __expf(sc[n] - gmax);
    sc[n] = p;
    lsum += p;
  }
  red[tid] = lsum; __syncthreads();
  for (int s = 128; s > 0; s >>= 1) {
    if (tid < s) red[tid] += red[tid + s];
    __syncthreads();
  }
  const float rinv = 1.0f / red[0];

  #pragma unroll
  for (int i = 0; i < NN / 256; ++i) {
    const int n = tid + 256 * i;
    att[b * NN + n] = sc[n] * rinv;
  }
}

// ---------------------------------------------------------------------------
// Kernel D: out = att @ xn  ([512,2048] x [2048,128]) via f32 WMMA.
// One wave per 16x16 tile; K=2048 -> 512 WMMAs (data is L2-resident).
// ---------------------------------------------------------------------------
__global__ __launch_bounds__(32) void attxn_wmma(const float* __restrict__ att,
                                                 const float* __restrict__ xn,
                                                 float* __restrict__ out) {
  const int rowTile = blockIdx.x;   // 0..31
  const int colTile = blockIdx.y;   // 0..7
  const int lane = threadIdx.x;
  const int half = lane >> 4;
  const int l16  = lane & 15;
  const int rowBase = rowTile * 16;
  const int colBase = colTile * 16;

  v8f acc = {};
  for (int k = 0; k < NN; k += 4) {
    const float* ap = att + (rowBase + l16) * NN + k + 2 * half;
    v2f a; a.x = ap[0]; a.y = ap[1];
    v2f b;
    b.x = xn[(k + 2 * half + 0) * INN + colBase + l16];
    b.y = xn[(k + 2 * half + 1) * INN + colBase + l16];
    acc = __builtin_amdgcn_wmma_f32_16x16x4_f32(false, a, false, b,
                                                (short)0, acc, false, false);
  }
  #pragma unroll
  for (int r = 0; r < 8; ++r)
    out[(rowBase + r + 8 * half) * INN + colBase + l16] = acc[r];
}

// ---------------------------------------------------------------------------
extern "C" void kernel_launch(void* const* d_in, const int* in_sizes, int n_in,
                              void* d_out, int out_size, void* d_ws, size_t ws_size,
                              hipStream_t stream) {
  const float* xo = (const float*)d_in[0];   // [512,128]
  const float* xn = (const float*)d_in[1];   // [2048,128]
  const float* W1 = (const float*)d_in[2];   // [256,64]
  const float* b1 = (const float*)d_in[3];   // [64]
  const float* W2 = (const float*)d_in[4];   // [64] (W2[:,0])
  // d_in[5] = b2: constant shift of scores -> softmax-invariant, dropped.

  float* st  = (float*)d_ws;                 // [2560,64] s rows then t rows
  float* uv  = st + (BB + NN) * HID;         // [2560]
  float* att = uv + (BB + NN);               // [512,2048]

  proj_wmma    <<<dim3(160, 4), 32,  0, stream>>>(xo, xn, W1, b1, st);
  rowdot_w2    <<<10,           256, 0, stream>>>(st, W2, uv);
  scores_softmax<<<BB,          256, 0, stream>>>(st, uv, W2, att);
  attxn_wmma   <<<dim3(32, 8),  32,  0, stream>>>(att, xn, (float*)d_out);
}